// QNetwork_30743375904783
// MI455X (gfx1250) — compile-verified
//
#include <hip/hip_runtime.h>
#include <hip/hip_bf16.h>

// ---- problem constants (match reference) ----
#define NPG 128
#define EPG 512
#define FN  32
#define FE  32
#define FG  16
#define HID 128
#define NA  8
#define HA  256

typedef __attribute__((ext_vector_type(2))) float v2f;
typedef __attribute__((ext_vector_type(8))) float v8f;

// Stage W [32x128] (row-major K x N) into LDS pre-swizzled into WMMA B-fragment
// layout: flat = ((kt*8 + nt)*32 + lane)*2 + j  holds  W[K = 4*kt + 2*(lane>>4) + j][N = 16*nt + (lane&15)]
__device__ __forceinline__ void fill_swizzled_B(float* sB, const float* __restrict__ W, int tid) {
#pragma unroll
    for (int it = 0; it < 16; ++it) {
        int f  = tid + it * 256;          // 4096 floats total
        int j  = f & 1;
        int l  = (f >> 1) & 31;
        int nt = (f >> 6) & 7;
        int kt = f >> 9;
        int h  = l >> 4;
        int c  = l & 15;
        sB[f] = W[(4 * kt + 2 * h + j) * HID + 16 * nt + c];
    }
}

// Fused GEMM [rows x 32] @ [32 x 128] + per-column (global-proj + bias) + ReLU + row-sum.
// Each wave processes M-tiles mt = wave, wave+8, ...; writes its 128 column partial sums
// to sWsum[wave*128 + n]. Uniform control flow per wave (WMMA needs EXEC all ones).
__device__ __forceinline__ void gemm_relu_colsum(const float* __restrict__ X, int mtiles,
                                                 const float* sB, const float* sGB,
                                                 float* sWsum, int tid) {
    const int wave = tid >> 5, lane = tid & 31;
    const int half = lane >> 4, col = lane & 15;

    float gb[8];
#pragma unroll
    for (int nt = 0; nt < 8; ++nt) gb[nt] = sGB[nt * 16 + col];

    float colsum[8] = {0.f, 0.f, 0.f, 0.f, 0.f, 0.f, 0.f, 0.f};

    for (int mt = wave; mt < mtiles; mt += 8) {
        // A fragment: lane holds A[M=col][K = 4*kt + 2*half + j]
        const float* Xr = X + (size_t)(mt * 16 + col) * 32 + 2 * half;
        v2f a[8];
#pragma unroll
        for (int kt = 0; kt < 8; ++kt)
            a[kt] = *(const v2f*)(Xr + 4 * kt);

#pragma unroll
        for (int nt = 0; nt < 8; ++nt) {
            v8f c = {0.f, 0.f, 0.f, 0.f, 0.f, 0.f, 0.f, 0.f};
#pragma unroll
            for (int kt = 0; kt < 8; ++kt) {
                v2f b = *(const v2f*)(sB + ((kt * 8 + nt) * 32 + lane) * 2);
                c = __builtin_amdgcn_wmma_f32_16x16x4_f32(
                        /*neg_a=*/false, a[kt], /*neg_b=*/false, b,
                        /*c_mod=*/(short)0, c, /*reuse_a=*/false, /*reuse_b=*/false);
            }
            // C layout: lane = column N = 16*nt+col; VGPR r = row M = r + 8*half
            float s = 0.f;
#pragma unroll
            for (int r = 0; r < 8; ++r) s += fmaxf(c[r] + gb[nt], 0.f);
            colsum[nt] += s;
        }
    }
    // combine row-halves (M 0..7 in lanes 0-15, M 8..15 in lanes 16-31)
#pragma unroll
    for (int nt = 0; nt < 8; ++nt)
        colsum[nt] += __shfl_xor(colsum[nt], 16, 32);
    if (lane < 16) {
#pragma unroll
        for (int nt = 0; nt < 8; ++nt)
            sWsum[wave * 128 + nt * 16 + lane] = colsum[nt];
    }
}

__global__ __launch_bounds__(256) void qnet_fused_kernel(
    const float* __restrict__ nodef, const float* __restrict__ edgef,
    const float* __restrict__ gfeat, const float* __restrict__ act,
    const float* __restrict__ W_ee, const float* __restrict__ W_eg, const float* __restrict__ b_e,
    const float* __restrict__ W_nn, const float* __restrict__ W_ng, const float* __restrict__ b_n,
    const float* __restrict__ W_gn, const float* __restrict__ W_ge, const float* __restrict__ b_g,
    const float* __restrict__ W_h,  const float* __restrict__ b_h,
    const float* __restrict__ W_o,  const float* __restrict__ b_o,
    float* __restrict__ out)
{
    __shared__ float sB[4096];      // swizzled weight tiles (16 KB, reused edge->node)
    __shared__ float sGBe[128];     // global_proj_edge + b_e
    __shared__ float sGBn[128];     // global_proj_node + b_n
    __shared__ float sEsum[128];    // per-graph edge activation column sums
    __shared__ float sWsum[8 * 128];// per-wave partial column sums
    __shared__ float sRed[256];     // reduction scratch

    const int g   = blockIdx.x;
    const int tid = threadIdx.x;

    // --- per-graph global projections: (g @ W_*g) + bias, 128 outputs each ---
    const float* gv = gfeat + (size_t)g * FG;
    if (tid < 128) {
        float acc = b_e[tid];
#pragma unroll
        for (int f = 0; f < FG; ++f) acc += gv[f] * W_eg[f * HID + tid];
        sGBe[tid] = acc;
    } else {
        const int n = tid - 128;
        float acc = b_n[n];
#pragma unroll
        for (int f = 0; f < FG; ++f) acc += gv[f] * W_ng[f * HID + n];
        sGBn[n] = acc;
    }

    // --- edge phase: 512 edges -> 32 M-tiles ---
    fill_swizzled_B(sB, W_ee, tid);
    __syncthreads();
    gemm_relu_colsum(edgef + (size_t)g * EPG * FE, 32, sB, sGBe, sWsum, tid);
    __syncthreads();

    if (tid < 128) {
        float s = 0.f;
#pragma unroll
        for (int w = 0; w < 8; ++w) s += sWsum[w * 128 + tid];
        sEsum[tid] = s;
    }
    // --- node phase: 128 nodes -> 8 M-tiles ---
    fill_swizzled_B(sB, W_nn, tid);
    __syncthreads();
    gemm_relu_colsum(nodef + (size_t)g * NPG * FN, 8, sB, sGBn, sWsum, tid);
    __syncthreads();

    // --- state value: n_mean @ W_gn + e_mean @ W_ge + b_g ---
    float p = 0.f;
    if (tid < 128) {
        float ns = 0.f;
#pragma unroll
        for (int w = 0; w < 8; ++w) ns += sWsum[w * 128 + tid];
        p = (ns * (1.0f / NPG)) * W_gn[tid] + (sEsum[tid] * (1.0f / EPG)) * W_ge[tid];
    }
    sRed[tid] = p;
    __syncthreads();
#pragma unroll
    for (int s = 128; s >= 1; s >>= 1) {
        if (tid < s) sRed[tid] += sRed[tid + s];
        __syncthreads();
    }
    const float sv = sRed[0] + b_g[0];
    __syncthreads();

    // --- action head: h = relu([sv, a] @ W_h + b_h); out = relu(h @ W_o + b_o) ---
    float hj = sv * W_h[tid] + b_h[tid];
    const float* av = act + (size_t)g * NA;
#pragma unroll
    for (int i = 0; i < NA; ++i) hj += av[i] * W_h[(1 + i) * HA + tid];
    hj = fmaxf(hj, 0.f);
    sRed[tid] = hj * W_o[tid];
    __syncthreads();
#pragma unroll
    for (int s = 128; s >= 1; s >>= 1) {
        if (tid < s) sRed[tid] += sRed[tid + s];
        __syncthreads();
    }
    if (tid == 0) out[g] = fmaxf(sRed[0] + b_o[0], 0.f);
}

extern "C" void kernel_launch(void* const* d_in, const int* in_sizes, int n_in,
                              void* d_out, int out_size, void* d_ws, size_t ws_size,
                              hipStream_t stream) {
    const float* nodef = (const float*)d_in[0];   // [N, FN]
    const float* edgef = (const float*)d_in[1];   // [E, FE]
    const float* gfeat = (const float*)d_in[2];   // [B, FG]
    const float* act   = (const float*)d_in[3];   // [B, NA]
    // d_in[4] node_graph, d_in[5] edge_graph: contiguous segments, unused
    const float* W_ee  = (const float*)d_in[6];
    const float* W_eg  = (const float*)d_in[7];
    const float* b_e   = (const float*)d_in[8];
    const float* W_nn  = (const float*)d_in[9];
    const float* W_ng  = (const float*)d_in[10];
    const float* b_n   = (const float*)d_in[11];
    const float* W_gn  = (const float*)d_in[12];
    const float* W_ge  = (const float*)d_in[13];
    const float* b_g   = (const float*)d_in[14];
    const float* W_h   = (const float*)d_in[15];
    const float* b_h   = (const float*)d_in[16];
    const float* W_o   = (const float*)d_in[17];
    const float* b_o   = (const float*)d_in[18];
    float* out = (float*)d_out;

    const int B = in_sizes[2] / FG;               // number of graphs (1024)

    qnet_fused_kernel<<<B, 256, 0, stream>>>(
        nodef, edgef, gfeat, act,
        W_ee, W_eg, b_e, W_nn, W_ng, b_n,
        W_gn, W_ge, b_g, W_h, b_h, W_o, b_o, out);
}